// MLP_RSNA11_73778948210892
// MI455X (gfx1250) — compile-verified
//
#include <hip/hip_runtime.h>

// CDNA5 / gfx1250: wave32, WMMA f32 16x16x4 (full-precision matrix op).
typedef __attribute__((ext_vector_type(2))) float v2f;
typedef __attribute__((ext_vector_type(8))) float v8f;

namespace {

constexpr int F     = 105;   // input features per row
constexpr int OUTC  = 75;    // output columns per row
constexpr int TROWS = 64;    // batch rows per block
constexpr int NTHR  = 128;   // 4 waves; each wave owns 16 rows
constexpr int NWAVE = NTHR / 32;
constexpr int GB    = 5;     // groups per WMMA burst (15 = 3*5, 10 = 2*5)

struct alignas(16) Smem {
  float xS[TROWS][F];          // staged input tile (coalesced b128 load)  26.9 KB
  float oS[TROWS][OUTC];       // staged output tile (coalesced b128 out)  19.2 KB
  float hS[NWAVE][GB][16][6];  // per-wave hidden transpose scratch         7.7 KB
  int   vS[75];                // scatter indices (sp|nf|ss concatenated)
  float w1S[3][18];            // W1 per branch, row-major [6][3]
  float b1S[3][6];
  float w2S[3][18];            // W2 per branch, row-major [C][6] (sp uses 6)
  float b2S[3][3];
};

// One branch: NG independent 3->6(relu)->C MLPs sharing weights.
// Layer 1, per group, is one V_WMMA_F32_16X16X4_F32:
//   D[m,n] = sum_k A[m,k]*B[k,n] + C[m,n]
//   m = batch row (16 per wave), k = feature (3 used, k=3 zero-padded),
//   n = hidden unit (6 used of 16); C fragment preloaded with b1 so D is
//   the pre-activation directly. Groups are processed in bursts of 5 so
//   the 5 WMMAs + LDS traffic pipeline behind a single dscnt wait.
template <int NG, int C, int VOFS, int BR>
__device__ __forceinline__ void run_branch(
    Smem& sm, const int* __restrict__ kIdx, int wave, int lane, int rb)
{
  constexpr int NB = NG / GB;
  const int n  = lane & 15;   // D column / A row index (M)
  const int hi = lane >> 4;   // 0: K=0,1 half   1: K=2,3 half (A layout)

  // B fragment: B[k,n] = W1[n][k] for k<3,n<6; zero padding elsewhere.
  // Assumed 32-bit B layout (mirror of documented A 16x4 layout):
  //   VGPR0: lanes0-15 -> B[0,n], lanes16-31 -> B[2,n]
  //   VGPR1: lanes0-15 -> B[1,n], lanes16-31 -> B[3,n] (= 0 pad)
  const float* W1 = sm.w1S[BR];
  v2f bfrag;
  bfrag.x = (n < 6) ? W1[n * 3 + (hi ? 2 : 0)] : 0.0f;
  bfrag.y = (n < 6 && hi == 0) ? W1[n * 3 + 1] : 0.0f;

  // C fragment: every D element of column n gets b1[n].
  const float bias = (n < 6) ? sm.b1S[BR][n] : 0.0f;
  const v8f cfrag = {bias, bias, bias, bias, bias, bias, bias, bias};

  for (int b = 0; b < NB; ++b) {
    const int g0 = b * GB;

    // ---- Gather 5 A fragments. k-indices are wave-uniform loads from
    //      global (constant-cache friendly); x comes from the LDS tile.
    //      A layout: lane row M = lane%16; lanes0-15 hold K0/K1,
    //      lanes16-31 hold K2/0.
    v2f a[GB];
#pragma unroll
    for (int j = 0; j < GB; ++j) {
      const int kb = 3 * (g0 + j);
      const int k0 = kIdx[kb + 0];
      const int k1 = kIdx[kb + 1];
      const int k2 = kIdx[kb + 2];
      const int kA = hi ? k2 : k0;
      a[j].x = sm.xS[rb + n][kA];
      a[j].y = hi ? 0.0f : sm.xS[rb + n][k1];
    }

    // ---- 5 back-to-back full-fp32 WMMAs ----
    v8f d[GB];
#pragma unroll
    for (int j = 0; j < GB; ++j)
      d[j] = __builtin_amdgcn_wmma_f32_16x16x4_f32(
          /*neg_a=*/false, a[j], /*neg_b=*/false, bfrag,
          /*c_mod=*/(short)0, cfrag, /*reuse_a=*/false, /*reuse_b=*/false);

    // ---- ReLU + transpose D-layout -> row-major h in wave-private LDS.
    //      D element (M = v + 8*hi, N = n) lives in d[j][v] of this lane.
    if (n < 6) {
#pragma unroll
      for (int j = 0; j < GB; ++j)
#pragma unroll
        for (int v = 0; v < 8; ++v) {
          const float hv = d[j][v];
          sm.hS[wave][j][v + 8 * hi][n] = hv > 0.0f ? hv : 0.0f;
        }
    }
    // LDS is in-order per wave; one wait for the whole burst before the
    // cross-lane reads below (asm also acts as a compiler barrier).
    asm volatile("s_wait_dscnt 0" ::: "memory");

    // ---- Layer 2 over the burst: 5*16*C tiny dot products, all lanes.
    for (int it = lane; it < GB * 16 * C; it += 32) {
      const int g   = it / (16 * C);
      const int rem = it - g * (16 * C);
      const int row = rem & 15;
      const int c   = rem >> 4;
      const float* w2 = &sm.w2S[BR][c * 6];
      float acc = sm.b2S[BR][c];
#pragma unroll
      for (int o = 0; o < 6; ++o)
        acc = fmaf(sm.hS[wave][g][row][o], w2[o], acc);
      sm.oS[rb + row][sm.vS[VOFS + (g0 + g) * C + c]] = acc;
    }
    asm volatile("" ::: "memory");  // keep hS reads before next burst's stores
  }
}

__global__ __launch_bounds__(NTHR) void fused_small_mlp(
    const float* __restrict__ x,
    const float* __restrict__ Wsp1, const float* __restrict__ bsp1,
    const float* __restrict__ Wsp2, const float* __restrict__ bsp2,
    const float* __restrict__ Wnf1, const float* __restrict__ bnf1,
    const float* __restrict__ Wnf2, const float* __restrict__ bnf2,
    const float* __restrict__ Wss1, const float* __restrict__ bss1,
    const float* __restrict__ Wss2, const float* __restrict__ bss2,
    const int* __restrict__ ksp, const int* __restrict__ vsp,
    const int* __restrict__ knf, const int* __restrict__ vnf,
    const int* __restrict__ kss, const int* __restrict__ vss,
    float* __restrict__ out, int nrows)
{
  __shared__ Smem sm;
  const int  t       = threadIdx.x;
  const long rowBase = (long)blockIdx.x * TROWS;

  // ---- Stage tiny weights / scatter table into LDS ----
  if (t < 75) sm.vS[t] = (t < 15) ? vsp[t] : (t < 45 ? vnf[t - 15] : vss[t - 45]);
  if (t < 18) {
    sm.w1S[0][t] = Wsp1[t]; sm.w1S[1][t] = Wnf1[t]; sm.w1S[2][t] = Wss1[t];
    sm.w2S[1][t] = Wnf2[t]; sm.w2S[2][t] = Wss2[t];
  }
  if (t < 6) {
    sm.b1S[0][t] = bsp1[t]; sm.b1S[1][t] = bnf1[t]; sm.b1S[2][t] = bss1[t];
    sm.w2S[0][t] = Wsp2[t];                 // spinal W2 is [1][6]
  }
  if (t < 3) { sm.b2S[1][t] = bnf2[t]; sm.b2S[2][t] = bss2[t]; }
  if (t == 0) sm.b2S[0][0] = bsp2[0];

  // ---- Stage input tile (coalesced, 128-bit) + zero output staging ----
  const bool   full = (rowBase + TROWS) <= (long)nrows;
  const float* xg   = x + rowBase * F;
  if (full) {
    const float4* src = reinterpret_cast<const float4*>(xg);
    float4*       dst = reinterpret_cast<float4*>(&sm.xS[0][0]);
    for (int i = t; i < TROWS * F / 4; i += NTHR) dst[i] = src[i];
  } else {
    for (int i = t; i < TROWS * F; i += NTHR) {
      const long r = rowBase + i / F;
      (&sm.xS[0][0])[i] = (r < nrows) ? xg[i] : 0.0f;
    }
  }
  {
    const float4  z = make_float4(0.0f, 0.0f, 0.0f, 0.0f);
    float4* dst = reinterpret_cast<float4*>(&sm.oS[0][0]);
    for (int i = t; i < TROWS * OUTC / 4; i += NTHR) dst[i] = z;
  }
  __syncthreads();

  // ---- Compute: each wave owns 16 consecutive rows of the tile ----
  const int wave = t >> 5;
  const int lane = t & 31;
  const int rb   = wave * 16;
  run_branch<15, 1, 0,  0>(sm, ksp, wave, lane, rb);  // spinal: 15 x (3->6->1)
  run_branch<10, 3, 15, 1>(sm, knf, wave, lane, rb);  // nfn:    10 x (3->6->3)
  run_branch<10, 3, 45, 2>(sm, kss, wave, lane, rb);  // ss:     10 x (3->6->3)
  __syncthreads();

  // ---- Write output tile back (coalesced, 128-bit) ----
  float* og = out + rowBase * OUTC;
  if (full) {
    const float4* src = reinterpret_cast<const float4*>(&sm.oS[0][0]);
    float4*       dst = reinterpret_cast<float4*>(og);
    for (int i = t; i < TROWS * OUTC / 4; i += NTHR) dst[i] = src[i];
  } else {
    for (int i = t; i < TROWS * OUTC; i += NTHR) {
      const long r = rowBase + i / OUTC;
      if (r < nrows) og[i] = (&sm.oS[0][0])[i];
    }
  }
}

} // namespace

extern "C" void kernel_launch(void* const* d_in, const int* in_sizes, int n_in,
                              void* d_out, int out_size, void* d_ws, size_t ws_size,
                              hipStream_t stream) {
  const float* x    = (const float*)d_in[0];
  const float* Wsp1 = (const float*)d_in[1];
  const float* bsp1 = (const float*)d_in[2];
  const float* Wsp2 = (const float*)d_in[3];
  const float* bsp2 = (const float*)d_in[4];
  const float* Wnf1 = (const float*)d_in[5];
  const float* bnf1 = (const float*)d_in[6];
  const float* Wnf2 = (const float*)d_in[7];
  const float* bnf2 = (const float*)d_in[8];
  const float* Wss1 = (const float*)d_in[9];
  const float* bss1 = (const float*)d_in[10];
  const float* Wss2 = (const float*)d_in[11];
  const float* bss2 = (const float*)d_in[12];
  const int*   ksp  = (const int*)d_in[13];
  const int*   vsp  = (const int*)d_in[14];
  const int*   knf  = (const int*)d_in[15];
  const int*   vnf  = (const int*)d_in[16];
  const int*   kss  = (const int*)d_in[17];
  const int*   vss  = (const int*)d_in[18];
  float*       out  = (float*)d_out;

  const int nrows = in_sizes[0] / F;                 // 262144
  const int grid  = (nrows + TROWS - 1) / TROWS;     // 4096 blocks

  fused_small_mlp<<<grid, NTHR, 0, stream>>>(
      x, Wsp1, bsp1, Wsp2, bsp2, Wnf1, bnf1, Wnf2, bnf2,
      Wss1, bss1, Wss2, bss2, ksp, vsp, knf, vnf, kss, vss,
      out, nrows);
}